// GraphConvLayer_8976481649176
// MI455X (gfx1250) — compile-verified
//
#include <hip/hip_runtime.h>

// GraphConv: out = relu( (X @ A_hat^T) @ W )  ==  relu( X @ M ),  M = A_hat^T @ W
// X:[65536,256] f32, A_hat:[256,256] f32, W:[256,64] f32, out:[65536,64] f32.
// Memory-bound: 64MB X read + 16MB out write @ 23.3 TB/s ~ 3.4us. f32 WMMA
// (v_wmma_f32_16x16x4_f32) delivers the 2.15 GFLOP at comparable time with
// exact fp32 precision, so no low-precision conversion is warranted.

typedef __attribute__((ext_vector_type(2))) float v2f;
typedef __attribute__((ext_vector_type(4))) float v4f;
typedef __attribute__((ext_vector_type(8))) float v8f;

#define C_DIM 256
#define D_DIM 64
#define N_ROWS 65536              // B*T
#define LDS_PITCH 260             // padded pitch: conflict-free b64 A-frag reads
#define TILE_ROWS 16
#define TILES_PER_BLOCK 4
#define LDS_TILE (TILE_ROWS * LDS_PITCH)   // 4160 floats = 16,640 B per buffer

// s_wait_asynccnt with a literal immediate (ASYNCcnt <= n)
#define WAIT_ASYNC(n) asm volatile("s_wait_asynccnt " #n ::: "memory")

// ---------------------------------------------------------------------------
// Kernel 1: M = A_hat^T @ W, written directly in WMMA B-fragment order:
//   ws[((kt*4 + nt)*32 + lane)*2 + j] = M[kt*4 + 2*(lane>>4) + j][nt*16 + (lane&15)]
// (32-bit B-matrix 4x16 layout: lanes 0-15 -> K pair 0,1; lanes 16-31 -> K pair 2,3)
// ---------------------------------------------------------------------------
__global__ __launch_bounds__(256) void prep_bfrag(const float* __restrict__ A,
                                                  const float* __restrict__ W,
                                                  float* __restrict__ wsB) {
    int gid  = blockIdx.x * 256 + threadIdx.x;    // 0..16383
    int j    = gid & 1;
    int lane = (gid >> 1) & 31;
    int frag = gid >> 6;                          // 0..255
    int nt   = frag & 3;
    int kt   = frag >> 2;                         // 0..63
    int K    = kt * 4 + 2 * (lane >> 4) + j;      // row of M (0..255)
    int N    = nt * 16 + (lane & 15);             // col of M (0..63)
    float acc = 0.0f;
    for (int c = 0; c < C_DIM; ++c)
        acc = __builtin_fmaf(A[c * C_DIM + K], W[c * D_DIM + N], acc);
    wsB[gid] = acc;
}

// ---------------------------------------------------------------------------
// Stage one 16-row x 256-col X tile (16 KB, contiguous in global) into LDS via
// the CDNA5 async path: 32x global_load_async_to_lds_b128 (512 B per op across
// the wave), GVS addressing (uniform SGPR base + lane*16 voffset + imm i*512).
// INST_OFFSET applies to both global and LDS addresses, so vdst only carries
// the per-row padding drift: lds_base + lane*16 + (i/2)*16.
// Tracked by ASYNCcnt; no VGPR round trip.
// ---------------------------------------------------------------------------
__device__ __forceinline__ void stage_tile_async(const float* __restrict__ g,
                                                 float* s, int lane) {
    const unsigned long long gbase = (unsigned long long)g;             // uniform
    const unsigned int voff = (unsigned int)lane * 16u;                 // per-lane
    const unsigned int lds_lane =
        (unsigned int)(unsigned long long)s + (unsigned int)lane * 16u; // per-lane
#pragma unroll
    for (int i = 0; i < 32; ++i) {
        const unsigned int ldsa = lds_lane + (unsigned int)(i >> 1) * 16u;
        asm volatile("global_load_async_to_lds_b128 %0, %1, %2 offset:%3"
                     :
                     : "v"(ldsa), "v"(voff), "s"(gbase), "i"(i * 512)
                     : "memory");
    }
}

// ---------------------------------------------------------------------------
// Consume one staged tile: 64 k-steps x 4 n-tiles of v_wmma_f32_16x16x4_f32.
// A-frag: conflict-free ds_load_b64 (pitch 260 -> banks {4m+k..4m+k+3} cover
// all 64). B-frags: coalesced 256B global_load_b64 bursts, WGP$-resident.
// ---------------------------------------------------------------------------
__device__ __forceinline__ void compute_store_tile(const float* sbuf,
                                                   const v2f* __restrict__ Bf,
                                                   float* __restrict__ op,
                                                   int lane) {
    const int h = lane >> 4;
    const int m = lane & 15;
    v8f acc0 = {}, acc1 = {}, acc2 = {}, acc3 = {};

    for (int kt = 0; kt < 64; ++kt) {
        v2f a  = *(const v2f*)(&sbuf[m * LDS_PITCH + kt * 4 + 2 * h]);
        v2f b0 = Bf[(kt * 4 + 0) * 32 + lane];
        v2f b1 = Bf[(kt * 4 + 1) * 32 + lane];
        v2f b2 = Bf[(kt * 4 + 2) * 32 + lane];
        v2f b3 = Bf[(kt * 4 + 3) * 32 + lane];
        acc0 = __builtin_amdgcn_wmma_f32_16x16x4_f32(false, a, false, b0, (short)0, acc0, false, false);
        acc1 = __builtin_amdgcn_wmma_f32_16x16x4_f32(false, a, false, b1, (short)0, acc1, false, false);
        acc2 = __builtin_amdgcn_wmma_f32_16x16x4_f32(false, a, false, b2, (short)0, acc2, false, false);
        acc3 = __builtin_amdgcn_wmma_f32_16x16x4_f32(false, a, false, b3, (short)0, acc3, false, false);
    }

    // D layout: VGPR v -> row (v + 8h); lane%16 -> column within each n-tile.
    float* o = op + (h * 8) * D_DIM + m;
#pragma unroll
    for (int v = 0; v < 8; ++v) {
        float* orow = o + v * D_DIM;
        orow[0]  = fmaxf(acc0[v], 0.0f);
        orow[16] = fmaxf(acc1[v], 0.0f);
        orow[32] = fmaxf(acc2[v], 0.0f);
        orow[48] = fmaxf(acc3[v], 0.0f);
    }
}

// ---------------------------------------------------------------------------
// Main kernel: one wave per block; 4 tiles per block, double-buffered LDS.
// Pipeline: prefetch tile t+1 asynchronously while tile t feeds the WMMAs;
// s_wait_asynccnt 32 releases tile t with tile t+1's 32 ops still in flight
// (async ops complete in order).
// ---------------------------------------------------------------------------
__global__ __launch_bounds__(32) void gcn_wmma(const float* __restrict__ X,
                                               const float* __restrict__ wsB,
                                               float* __restrict__ out) {
    __shared__ float sX[2][LDS_TILE];             // 33,280 B
    const int lane = threadIdx.x;                 // 0..31 (wave32)
    const long t0 = (long)blockIdx.x * TILES_PER_BLOCK;
    const v2f* Bf = (const v2f*)wsB;

    stage_tile_async(X + t0 * (TILE_ROWS * C_DIM), sX[0], lane);

    for (int t = 0; t < TILES_PER_BLOCK; ++t) {
        if (t + 1 < TILES_PER_BLOCK) {
            stage_tile_async(X + (t0 + t + 1) * (long)(TILE_ROWS * C_DIM),
                             sX[(t + 1) & 1], lane);
            WAIT_ASYNC(32);                       // tile t complete; t+1 in flight
        } else {
            WAIT_ASYNC(0);                        // drain
        }
        __syncthreads();                          // ordering fence (1 wave: waits only)
        compute_store_tile(sX[t & 1], Bf,
                           out + (t0 + t) * (long)(TILE_ROWS * D_DIM), lane);
    }
}

extern "C" void kernel_launch(void* const* d_in, const int* in_sizes, int n_in,
                              void* d_out, int out_size, void* d_ws, size_t ws_size,
                              hipStream_t stream) {
    const float* x  = (const float*)d_in[0];      // [B,T,C,1] f32
    const float* Ah = (const float*)d_in[1];      // [C,C] f32
    const float* W  = (const float*)d_in[2];      // [C,D] f32
    float* outp = (float*)d_out;                  // [B,T,D,1] f32
    float* wsB  = (float*)d_ws;                   // 64 KB B-fragment staging

    prep_bfrag<<<64, 256, 0, stream>>>(Ah, W, wsB);
    gcn_wmma<<<N_ROWS / (TILE_ROWS * TILES_PER_BLOCK), 32, 0, stream>>>(x, wsB, outp);
}